// AutoconstraintModel_87153476370861
// MI455X (gfx1250) — compile-verified
//
#include <hip/hip_runtime.h>

typedef float v2f __attribute__((ext_vector_type(2)));
typedef float v8f __attribute__((ext_vector_type(8)));

#define D   128
#define NPG 64

// ---------------------------------------------------------------------------
// Kernel 1: edge_partner logits.
// Grid: B/2 blocks x 256 threads (8 wave32). Block handles 2 graphs.
// Wave w: graph (w>>2), M-tile (w&3)  -> 16 node rows, all 128 hidden cols.
// H = node @ Wp1[128:256];  logit = relu(H + base) . Wp2 + bp2
// base[g] = cur_g @ Wp1[0:128] + glob_g @ Wp1[256:384] + bp1   (per graph)
// ---------------------------------------------------------------------------
__global__ __launch_bounds__(256) void partner_kernel(
    const float* __restrict__ node, const float* __restrict__ glob,
    const float* __restrict__ Wp1,  const float* __restrict__ bp1,
    const float* __restrict__ Wp2,  const float* __restrict__ bp2,
    float* __restrict__ out)
{
  extern __shared__ float smem[];
  float* lds_wb   = smem;                 // 16384 f : swizzled Wp1[128:256]
  float* lds_base = smem + 16384;         //   256 f : base vectors (2 graphs)
  float* lds_vec  = smem + 16384 + 256;   //   512 f : cur/glob staging
  float* lds_wp2  = smem + 16384 + 768;   //   128 f

  const int tid = threadIdx.x;
  const int g0  = blockIdx.x * 2;

  // Stage middle block of Wp1 into LDS, pair-interleaved + rotated:
  // (k,n): p=k>>1, col=(n + 16*(p&1)) & 127 -> lds[p*256 + col*2 + (k&1)]
  for (int i = tid; i < D * D; i += 256) {
    int k = i >> 7, n = i & 127;
    float v = Wp1[(D + k) * D + n];
    int p = k >> 1;
    int col = (n + ((p & 1) << 4)) & 127;
    lds_wb[p * 256 + col * 2 + (k & 1)] = v;
  }
  {   // stage cur / glob vectors for both graphs
    int g = tid >> 7, j = tid & 127;
    int graph = g0 + g;
    lds_vec[(g * 2 + 0) * D + j] = node[(graph * NPG + NPG - 1) * D + j];
    lds_vec[(g * 2 + 1) * D + j] = glob[graph * D + j];
  }
  if (tid < D) lds_wp2[tid] = Wp2[tid];
  __syncthreads();

  {   // base[g][j] with VALU (tiny)
    int g = tid >> 7, j = tid & 127;
    float s = bp1[j];
    const float* cv = &lds_vec[(g * 2 + 0) * D];
    const float* gv = &lds_vec[(g * 2 + 1) * D];
#pragma unroll 8
    for (int k = 0; k < D; ++k) {
      s += cv[k] * Wp1[k * D + j];
      s += gv[k] * Wp1[(2 * D + k) * D + j];
    }
    lds_base[g * D + j] = s;
  }
  __syncthreads();

  const int wave = tid >> 5, lane = tid & 31;
  const int half = lane >> 4, l16 = lane & 15;
  const int g = wave >> 2, mtile = wave & 3;
  const int graph = g0 + g;
  const int row = graph * NPG + mtile * 16 + l16;
  const float* rowp = node + (size_t)row * D + 2 * half;

  // A fragments for entire K=128 held in registers (32 x v2f = 64 VGPRs)
  v2f af[32];
#pragma unroll
  for (int s = 0; s < 32; ++s) af[s] = *(const v2f*)(rowp + 4 * s);

  const v2f* wb2 = (const v2f*)lds_wb;
  const int coln0 = l16;                       // half==0 column
  const int coln1 = (l16 + 16) & 127;          // rotation applied per p parity

  float acc[8];
#pragma unroll
  for (int r = 0; r < 8; ++r) acc[r] = 0.f;

  for (int nt = 0; nt < 8; ++nt) {
    int n = nt * 16 + l16;
    int col = ((n + 16 * half) & 127);
    v8f c = {};
#pragma unroll
    for (int s = 0; s < 32; ++s) {
      v2f b = wb2[(2 * s + half) * D + col];
      c = __builtin_amdgcn_wmma_f32_16x16x4_f32(false, af[s], false, b,
                                                (short)0, c, false, false);
    }
    float bn = lds_base[g * D + n];
    float wn = lds_wp2[n];
#pragma unroll
    for (int r = 0; r < 8; ++r) {
      float v = c[r] + bn;
      v = v > 0.f ? v : 0.f;
      acc[r] += v * wn;
    }
  }
  // reduce over the 16 lanes of each half (masks <16 stay within halves)
#pragma unroll
  for (int r = 0; r < 8; ++r) {
    float a = acc[r];
    a += __shfl_xor(a, 1, 32);
    a += __shfl_xor(a, 2, 32);
    a += __shfl_xor(a, 4, 32);
    a += __shfl_xor(a, 8, 32);
    acc[r] = a;
  }
  if (l16 == 0) {
    float b2 = bp2[0];
#pragma unroll
    for (int r = 0; r < 8; ++r)
      out[graph * NPG + mtile * 16 + r + 8 * half] = acc[r] + b2;
  }
}

// ---------------------------------------------------------------------------
// Kernel 2: edge_label logits.  Grid: B/128 blocks x 256 threads.
// Wave handles 16 B-rows; 3 chained WMMA GEMMs (384->128->128->14),
// hidden activations transposed through LDS between layers.
// ---------------------------------------------------------------------------
__global__ __launch_bounds__(256) void label_kernel(
    const float* __restrict__ node, const float* __restrict__ glob,
    const float* __restrict__ Wl1,  const float* __restrict__ bl1,
    const float* __restrict__ Wl2,  const float* __restrict__ bl2,
    const float* __restrict__ Wl3,  const float* __restrict__ bl3,
    const int* __restrict__ pgi,    const int* __restrict__ pni,
    float* __restrict__ out, int outoff)
{
  extern __shared__ float smem[];
  float* lds_h1 = smem;                      // 8*16*128 = 16384 f
  float* lds_h2 = smem + 8 * 16 * D;         // 16384 f
  float* lds_w3 = smem + 2 * 8 * 16 * D;     // 128*16 zero-padded Wl3

  const int tid = threadIdx.x;
  for (int i = tid; i < D * 16; i += 256) {
    int k = i >> 4, n = i & 15;
    lds_w3[i] = (n < 14) ? Wl3[k * 14 + n] : 0.f;
  }

  const int wave = tid >> 5, lane = tid & 31;
  const int half = lane >> 4, l16 = lane & 15;
  const int rowid = blockIdx.x * 128 + wave * 16 + l16;
  const int pg = pgi[rowid];
  const int pn = pni[rowid];
  const float* src0 = node + ((size_t)(pg + 1) * NPG - 1) * D; // current
  const float* src1 = node + (size_t)pn * D;                   // partner
  const float* src2 = glob + (size_t)pg * D;                   // global

  // ---- layer 1: [16,384] @ [384,128] ----
  for (int nt = 0; nt < 8; ++nt) {
    int n = nt * 16 + l16;
    v8f c = {};
#pragma unroll
    for (int seg = 0; seg < 3; ++seg) {
      const float* ap = (seg == 0 ? src0 : seg == 1 ? src1 : src2) + 2 * half;
      const float* bp = Wl1 + (size_t)(seg * D + 2 * half) * D + n;
#pragma unroll
      for (int s = 0; s < 32; ++s) {
        v2f a = *(const v2f*)(ap + 4 * s);
        v2f b;
        b.x = bp[(size_t)(4 * s) * D];
        b.y = bp[(size_t)(4 * s + 1) * D];
        c = __builtin_amdgcn_wmma_f32_16x16x4_f32(false, a, false, b,
                                                  (short)0, c, false, false);
      }
    }
    float bv = bl1[n];
#pragma unroll
    for (int r = 0; r < 8; ++r) {
      float v = c[r] + bv;
      lds_h1[(wave * 16 + r + 8 * half) * D + n] = v > 0.f ? v : 0.f;
    }
  }
  __syncthreads();

  // ---- layer 2: [16,128] @ [128,128] ----
  for (int nt = 0; nt < 8; ++nt) {
    int n = nt * 16 + l16;
    v8f c = {};
    const float* ap = lds_h1 + (wave * 16 + l16) * D + 2 * half;
    const float* bp = Wl2 + (size_t)(2 * half) * D + n;
#pragma unroll
    for (int s = 0; s < 32; ++s) {
      v2f a = *(const v2f*)(ap + 4 * s);
      v2f b;
      b.x = bp[(size_t)(4 * s) * D];
      b.y = bp[(size_t)(4 * s + 1) * D];
      c = __builtin_amdgcn_wmma_f32_16x16x4_f32(false, a, false, b,
                                                (short)0, c, false, false);
    }
    float bv = bl2[n];
#pragma unroll
    for (int r = 0; r < 8; ++r) {
      float v = c[r] + bv;
      lds_h2[(wave * 16 + r + 8 * half) * D + n] = v > 0.f ? v : 0.f;
    }
  }
  __syncthreads();

  // ---- layer 3: [16,128] @ [128,16(pad of 14)] ----
  {
    v8f c = {};
    const float* ap = lds_h2 + (wave * 16 + l16) * D + 2 * half;
#pragma unroll
    for (int s = 0; s < 32; ++s) {
      v2f a = *(const v2f*)(ap + 4 * s);
      v2f b;
      b.x = lds_w3[(4 * s + 2 * half) * 16 + l16];
      b.y = lds_w3[(4 * s + 2 * half + 1) * 16 + l16];
      c = __builtin_amdgcn_wmma_f32_16x16x4_f32(false, a, false, b,
                                                (short)0, c, false, false);
    }
    if (l16 < 14) {
      float bv = bl3[l16];
#pragma unroll
      for (int r = 0; r < 8; ++r) {
        int grow = blockIdx.x * 128 + wave * 16 + r + 8 * half;
        out[outoff + grow * 14 + l16] = c[r] + bv;
      }
    }
  }
}

extern "C" void kernel_launch(void* const* d_in, const int* in_sizes, int n_in,
                              void* d_out, int out_size, void* d_ws, size_t ws_size,
                              hipStream_t stream) {
  (void)n_in; (void)out_size; (void)d_ws; (void)ws_size;
  const float* node = (const float*)d_in[0];
  const float* glob = (const float*)d_in[1];
  const float* Wp1  = (const float*)d_in[2];
  const float* bp1  = (const float*)d_in[3];
  const float* Wp2  = (const float*)d_in[4];
  const float* bp2  = (const float*)d_in[5];
  const float* Wl1  = (const float*)d_in[6];
  const float* bl1  = (const float*)d_in[7];
  const float* Wl2  = (const float*)d_in[8];
  const float* bl2  = (const float*)d_in[9];
  const float* Wl3  = (const float*)d_in[10];
  const float* bl3  = (const float*)d_in[11];
  const int*   pgi  = (const int*)d_in[12];
  const int*   pni  = (const int*)d_in[13];
  float* out = (float*)d_out;

  const int B = in_sizes[1] / D;   // 8192 graphs
  const int N = in_sizes[0] / D;   // 524288 nodes

  const size_t smem1 = (16384 + 256 + 512 + 128) * sizeof(float);      // ~68 KB
  const size_t smem2 = (2 * 8 * 16 * D + D * 16) * sizeof(float);      // ~136 KB

  partner_kernel<<<B / 2, 256, smem1, stream>>>(node, glob, Wp1, bp1, Wp2, bp2, out);
  label_kernel<<<B / 128, 256, smem2, stream>>>(node, glob, Wl1, bl1, Wl2, bl2,
                                                Wl3, bl3, pgi, pni, out, N);
}